// RNN_MultiRegional_D1D2_79285096284298
// MI455X (gfx1250) — compile-verified
//
#include <hip/hip_runtime.h>
#include <hip/hip_bf16.h>
#include <stddef.h>

// ---------------------------------------------------------------------------
// Multi-regional RNN rollout for MI455X (gfx1250).
//   drive^T[i][b] = sum_j W[i][j] * hT[j][b]  via V_WMMA_F32_16X16X4_F32.
//   W pre-swizzled into WMMA A-fragment order (one b64 per WMMA).
//   hT B-slices staged into LDS with GLOBAL_LOAD_ASYNC_TO_LDS_B128
//   (ASYNCcnt-tracked, double buffered) and shared by 4 waves per block.
// ---------------------------------------------------------------------------

#define HID    512
#define HH     256
#define INPN   1
#define OUTN   1
#define EXC    359
#define NREAL  1793
#define ALM0   1280
#define BATCH  128
#define TSTEPS 300
#define NP     1856           // padded N: 116 tiles of 16
#define K4T    (NP / 4)       // 464 K-steps of 4
#define MTILES (NP / 16)      // 116
#define NTILES (BATCH / 16)   // 8
#define KCHUNK 64             // K rows staged in LDS per double-buffer slot
#define NCHUNKS (NP / KCHUNK) // 29

typedef float v2f __attribute__((ext_vector_type(2)));
typedef float v8f __attribute__((ext_vector_type(8)));
typedef int   v4i __attribute__((ext_vector_type(4)));
typedef __attribute__((address_space(1))) v4i* glb_v4i_p;
typedef __attribute__((address_space(3))) v4i* lds_v4i_p;

#if defined(__has_builtin)
#if __has_builtin(__builtin_amdgcn_global_load_async_to_lds_b128)
#define HAVE_ASYNC_LDS 1
#endif
#endif

#if defined(HAVE_ASYNC_LDS)
#if __has_builtin(__builtin_amdgcn_s_wait_asynccnt)
#define WAIT_ASYNC(n) __builtin_amdgcn_s_wait_asynccnt(n)
#else
#define WAIT_ASYNC(n) asm volatile("s_wait_asynccnt %0" ::"n"(n))
#endif
#else
#define WAIT_ASYNC(n) ((void)0)
#endif

__device__ __forceinline__ float clipw(float w) {
    return fminf(fmaxf(w, 1e-10f), 1.0f);    // F.hardtanh(w, 1e-10, 1)
}

// Deterministic ~10%-density mask approximating torch nn.init.sparse_(.., 0.9)
// (numpy PCG64 permutation not reproducible on-device; compile-only loop).
__device__ __forceinline__ float spmask(int r, int c) {
    unsigned h = (unsigned)r * 2654435761u ^ ((unsigned)c * 40503u + 0x9E3779B9u);
    h ^= h >> 13; h *= 0x5bd1e995u; h ^= h >> 15;
    return (h % 10u == 0u) ? 1.0f : 0.0f;
}

// W[i][j] of the assembled connectivity matrix (reference _build_W).
__device__ float w_val(int i, int j,
    const float* d12d1, const float* d12d2, const float* d22d1, const float* d22d2,
    const float* d22stn, const float* d12thal, const float* stn2thal,
    const float* thal2alm, const float* alm2alm, const float* thal2d1,
    const float* thal2d2, const float* alm2d1, const float* alm2d2,
    const float* inpw1, const float* inpw2)
{
    if (i < HH) {                                   // D1 rows
        if (j < HH)        return -(spmask(i, j) * clipw(d12d1[i * HH + j]));
        if (j < 2 * HH)    return -(spmask(i, j - HH) * clipw(d22d1[i * HH + (j - HH)]));
        if (j < 3 * HH)    return 0.0f;
        if (j < 3 * HH + HID)
            return (i < HH / 2) ? clipw(thal2d1[i * HID + (j - 3 * HH)]) : 0.0f;
        if (j < NREAL - INPN) {
            int jc = j - ALM0;
            return (i >= HH / 2 && jc < EXC) ? clipw(alm2d1[i * HID + jc]) : 0.0f;
        }
        return (i < HH / 2) ? clipw(inpw1[i]) : 0.0f;
    }
    if (i < 2 * HH) {                               // D2 rows
        int r = i - HH;
        if (j < HH)        return -(spmask(r, j) * clipw(d12d2[r * HH + j]));
        if (j < 2 * HH)    return -(spmask(r, j - HH) * clipw(d22d2[r * HH + (j - HH)]));
        if (j < 3 * HH)    return 0.0f;
        if (j < 3 * HH + HID) return clipw(thal2d2[r * HID + (j - 3 * HH)]);
        if (j < NREAL - INPN) {
            int jc = j - ALM0;
            return (jc < EXC) ? clipw(alm2d2[r * HID + jc]) : 0.0f;
        }
        return clipw(inpw2[r]);
    }
    if (i < 3 * HH) {                               // STN rows
        int r = i - 2 * HH;
        if (j >= HH && j < 2 * HH) return clipw(d22stn[r * HH + (j - HH)]);
        return 0.0f;
    }
    if (i < 3 * HH + HID) {                         // thal rows
        int r = i - 3 * HH;
        if (j < HH) return clipw(d12thal[r * HH + j]);
        if (j >= 2 * HH && j < 3 * HH) return -clipw(stn2thal[r * HH + (j - 2 * HH)]);
        return 0.0f;
    }
    if (i < NREAL - INPN) {                         // ALM rows
        int r = i - ALM0;
        if (j >= 3 * HH && j < ALM0) return clipw(thal2alm[r * HID + (j - 3 * HH)]);
        if (j >= ALM0 && j < NREAL - INPN) {
            int jc = j - ALM0;
            float s = (jc < EXC) ? 1.0f : -1.0f;    // Dale sign
            return clipw(alm2alm[r * HID + jc]) * s;
        }
        return 0.0f;
    }
    return 0.0f;                                    // ITI row
}

// Build W padded to NP x NP in WMMA A-fragment order:
// tile (mtile, k4): lane = m%16 + 16*(kk>>1), vgpr r = kk&1
// (A 16x4 layout: v0 = K{0,2}, v1 = K{1,3}).
__global__ void build_wswz_kernel(
    const float* d12d1, const float* d12d2, const float* d22d1, const float* d22d2,
    const float* d22stn, const float* d12thal, const float* stn2thal,
    const float* thal2alm, const float* alm2alm, const float* thal2d1,
    const float* thal2d2, const float* alm2d1, const float* alm2d2,
    const float* inpw1, const float* inpw2, float* __restrict__ wswz)
{
    size_t idx = (size_t)blockIdx.x * blockDim.x + threadIdx.x;
    if (idx >= (size_t)NP * NP) return;
    int i = (int)(idx / NP);
    int j = (int)(idx % NP);
    float w = 0.0f;
    if (i < NREAL && j < NREAL)
        w = w_val(i, j, d12d1, d12d2, d22d1, d22d2, d22stn, d12thal, stn2thal,
                  thal2alm, alm2alm, thal2d1, thal2d2, alm2d1, alm2d2, inpw1, inpw2);
    int mtile = i >> 4, ml = i & 15;
    int k4 = j >> 2, kk = j & 3;
    int lane = ml + ((kk >> 1) << 4);
    int r = kk & 1;
    wswz[(((size_t)mtile * K4T + k4) * 32 + lane) * 2 + r] = w;
}

// Transpose-pad initial state: hT[j][b] = hn[0][b][j]
__global__ void init_state_kernel(const float* __restrict__ hn,
                                  const float* __restrict__ xn,
                                  float* __restrict__ hT, float* __restrict__ xT)
{
    int idx = blockIdx.x * blockDim.x + threadIdx.x;
    if (idx >= NP * BATCH) return;
    int j = idx / BATCH, b = idx % BATCH;
    float h = 0.0f, x = 0.0f;
    if (j < NREAL) { h = hn[(size_t)b * NREAL + j]; x = xn[(size_t)b * NREAL + j]; }
    hT[idx] = h; xT[idx] = x;
}

// Stage one KCHUNK x 16 slice of hT into LDS (async if available).
// 64 rows * 64B = 4KB = 256 b128 transfers; 128 threads -> 2 each.
__device__ __forceinline__ void stage_chunk(const float* __restrict__ hT,
                                            int nt0, int c, float* lbuf, int tid)
{
#pragma unroll
    for (int i = 0; i < 2; ++i) {
        int flat = tid + 128 * i;
        int row = flat >> 2, seg = flat & 3;
        const float* g = hT + ((size_t)(c * KCHUNK + row)) * BATCH + nt0 + seg * 4;
        float* l = lbuf + row * 16 + seg * 4;
#if defined(HAVE_ASYNC_LDS)
        __builtin_amdgcn_global_load_async_to_lds_b128(
            (glb_v4i_p)(void*)(const_cast<float*>(g)),
            (lds_v4i_p)(void*)l, 0, 0);
#else
        *(float4*)l = *(const float4*)g;
#endif
    }
}

// One RNN timestep. Block = 4 waves (128 thr): wave w owns M-tile mgroup*4+w,
// all waves share batch-tile ntile and the LDS-staged B slice.
__global__ __launch_bounds__(128) void step_kernel(
    const float* __restrict__ wswz, const float* __restrict__ hT_in,
    const float* __restrict__ xT_in, float* __restrict__ hT_out,
    float* __restrict__ xT_out, const float* __restrict__ inp,
    const float* __restrict__ cue, const float* __restrict__ inhib,
    const float* __restrict__ noise, float* __restrict__ hs,
    float* __restrict__ xs, int t)
{
    __shared__ float ldsB[2][KCHUNK * 16];

    const int tid   = threadIdx.x;
    const int lane  = tid & 31;
    const int wv    = tid >> 5;
    const int mgrp  = blockIdx.x >> 3;
    const int ntile = blockIdx.x & 7;
    const int mtile = mgrp * 4 + wv;
    const int nt0   = ntile * 16;

    v8f acc = {0.f, 0.f, 0.f, 0.f, 0.f, 0.f, 0.f, 0.f};

    const float* aptr  = wswz + (size_t)mtile * K4T * 64 + (size_t)lane * 2;
    const int    khigh = (lane >> 4) * 2;     // low lanes K{0,1}, high K{2,3}
    const int    bl    = lane & 15;

    stage_chunk(hT_in, nt0, 0, &ldsB[0][0], tid);

    for (int c = 0; c < NCHUNKS; ++c) {
        if (c + 1 < NCHUNKS) {
            stage_chunk(hT_in, nt0, c + 1, &ldsB[(c + 1) & 1][0], tid);
            WAIT_ASYNC(2);                    // chunk c complete (next still in flight)
        } else {
            WAIT_ASYNC(0);
        }
        __syncthreads();

        const float* lb = &ldsB[c & 1][0];
        __builtin_prefetch(aptr + (KCHUNK / 4) * 64, 0, 0);   // A stream ahead
#pragma unroll
        for (int kk = 0; kk < KCHUNK / 4; ++kk) {
            v2f a = *(const v2f*)(aptr + (size_t)kk * 64);
            int kb = kk * 4 + khigh;
            v2f b;
            b.x = lb[(kb + 0) * 16 + bl];
            b.y = lb[(kb + 1) * 16 + bl];
            acc = __builtin_amdgcn_wmma_f32_16x16x4_f32(
                false, a, false, b, (short)0, acc, false, false);
        }
        aptr += (KCHUNK / 4) * 64;
        __syncthreads();
    }

    // Epilogue: D layout — vgpr r: lanes0-15 -> M=r, lanes16-31 -> M=r+8.
    const int b = nt0 + bl;
#pragma unroll
    for (int r = 0; r < 8; ++r) {
        int i = mtile * 16 + r + ((lane >> 4) << 3);
        float drive = acc[r];
        if (i < NREAL) {
            size_t off = ((size_t)b * TSTEPS + t) * NREAL + i;
            float add = inhib[off] + 0.01f * noise[off];
            if (i < NREAL - INPN) add += 0.1f;                      // tonic
            else                  add += inp[b * TSTEPS + t] + cue[b * TSTEPS + t];
            float xp = xT_in[(size_t)i * BATCH + b];
            float xv = xp + 0.1f * (-xp + drive + add);
            float hv = fmaxf(xv, 0.0f);
            xT_out[(size_t)i * BATCH + b] = xv;
            hT_out[(size_t)i * BATCH + b] = hv;
            xs[off] = xv;
            hs[off] = hv;
        } else {
            xT_out[(size_t)i * BATCH + b] = 0.0f;
            hT_out[(size_t)i * BATCH + b] = 0.0f;
        }
    }
}

// outs[b,t] = hs[b,t,ALM0:ALM0+EXC] . out_w
__global__ void out_kernel(const float* __restrict__ hs,
                           const float* __restrict__ ow,
                           float* __restrict__ outs)
{
    int idx = blockIdx.x * blockDim.x + threadIdx.x;   // b*T + t
    if (idx >= BATCH * TSTEPS) return;
    const float* hrow = hs + (size_t)idx * NREAL + ALM0;
    float s = 0.0f;
    for (int e = 0; e < EXC; ++e) s += hrow[e] * ow[e];
    outs[idx] = s;
}

extern "C" void kernel_launch(void* const* d_in, const int* in_sizes, int n_in,
                              void* d_out, int out_size, void* d_ws, size_t ws_size,
                              hipStream_t stream) {
    const float* inp   = (const float*)d_in[0];
    const float* cue   = (const float*)d_in[1];
    const float* inhib = (const float*)d_in[2];
    const float* hn    = (const float*)d_in[3];
    const float* xn    = (const float*)d_in[4];
    const float* noise = (const float*)d_in[5];
    const float* d12d1   = (const float*)d_in[6];
    const float* d12d2   = (const float*)d_in[7];
    const float* d22d1   = (const float*)d_in[8];
    const float* d22d2   = (const float*)d_in[9];
    const float* d22stn  = (const float*)d_in[10];
    const float* d12thal = (const float*)d_in[11];
    const float* stn2thal= (const float*)d_in[12];
    const float* thal2alm= (const float*)d_in[13];
    const float* alm2alm = (const float*)d_in[14];
    const float* thal2d1 = (const float*)d_in[15];
    const float* thal2d2 = (const float*)d_in[16];
    const float* alm2d1  = (const float*)d_in[17];
    const float* alm2d2  = (const float*)d_in[18];
    const float* inpw1   = (const float*)d_in[19];
    const float* inpw2   = (const float*)d_in[20];
    const float* outw    = (const float*)d_in[21];

    float* ws   = (float*)d_ws;
    float* wswz = ws;                               // NP*NP
    float* hTa  = wswz + (size_t)NP * NP;           // NP*BATCH each
    float* hTb  = hTa + (size_t)NP * BATCH;
    float* xTa  = hTb + (size_t)NP * BATCH;
    float* xTb  = xTa + (size_t)NP * BATCH;

    const size_t BTN = (size_t)BATCH * TSTEPS * NREAL;
    float* hs   = (float*)d_out;
    float* xs   = hs + BTN;
    float* outs = xs + BTN;

    {
        size_t total = (size_t)NP * NP;
        dim3 g((unsigned)((total + 255) / 256));
        build_wswz_kernel<<<g, 256, 0, stream>>>(
            d12d1, d12d2, d22d1, d22d2, d22stn, d12thal, stn2thal,
            thal2alm, alm2alm, thal2d1, thal2d2, alm2d1, alm2d2,
            inpw1, inpw2, wswz);
    }
    init_state_kernel<<<(NP * BATCH + 255) / 256, 256, 0, stream>>>(hn, xn, hTa, xTa);

    for (int t = 0; t < TSTEPS; ++t) {
        const float* hi = (t & 1) ? hTb : hTa;
        const float* xi = (t & 1) ? xTb : xTa;
        float* ho = (t & 1) ? hTa : hTb;
        float* xo = (t & 1) ? xTa : xTb;
        step_kernel<<<(MTILES / 4) * NTILES, 128, 0, stream>>>(
            wswz, hi, xi, ho, xo, inp, cue, inhib, noise, hs, xs, t);
    }

    out_kernel<<<(BATCH * TSTEPS + 255) / 256, 256, 0, stream>>>(hs, outw, outs);
}